// PowerSpectrum_86088324481926
// MI455X (gfx1250) — compile-verified
//
#include <hip/hip_runtime.h>
#include <hip/hip_bf16.h>

typedef __attribute__((ext_vector_type(2))) float v2f;
typedef __attribute__((ext_vector_type(8))) float v8f;

// D(16x16) += A(16x4) * B(4x16), f32, via V_WMMA_F32_16X16X4_F32.
// 32-bit A 16x4 layout: lanes 0-15: M=lane, v0=K0, v1=K1; lanes 16-31: M=lane-16, v0=K2, v1=K3.
// B 4x16 layout (rows striped across lanes): v0: lanes0-15 K=0 | lanes16-31 K=2; v1: K=1 | K=3.
// Both A and B gather as c[k*128 + colbase + (lane&15)] with k = 2*half (+1 for .y).

// Masked gather without exec divergence: clamp the address (always in bounds,
// KK >= 1), select the value. Compile-time KK folds most masks away.
template <int KK>
__device__ __forceinline__ float ldmask(const float* __restrict__ c, int k, int col) {
    const int kc = (k < KK) ? k : (KK - 1);
    const float v = c[kc * 128 + col];
    return (k < KK) ? v : 0.0f;
}

template <int KK>
__device__ __forceinline__ void gram_tile_row(const float* __restrict__ c,
                                              float* __restrict__ ob,
                                              const float scale,
                                              const int I, const int lc,
                                              const int half) {
    const int k0 = 2 * half;      // K index held in VGPR .x
    const int k1 = k0 + 1;        // K index held in VGPR .y

    // A tile for this 16-row block (reused across all 8 J tiles)
    v2f a0, a1;
    a0.x = ldmask<KK>(c, k0, I + lc);
    a0.y = ldmask<KK>(c, k1, I + lc);
    if (KK > 4) {
        a1.x = ldmask<KK>(c, k0 + 4, I + lc);
        a1.y = ldmask<KK>(c, k1 + 4, I + lc);
    }

    // Unroll 2 (not 8): keeps VGPR pressure well under 256 so multiple blocks
    // co-reside per WGP -> more waves feeding the NT store queues. The kernel
    // is store-bandwidth-bound; wave count beats per-wave ILP here.
    #pragma unroll 2
    for (int tJ = 0; tJ < 8; ++tJ) {
        const int J = tJ << 4;

        v2f b0;
        b0.x = ldmask<KK>(c, k0, J + lc);
        b0.y = ldmask<KK>(c, k1, J + lc);

        v8f acc = {};
        acc = __builtin_amdgcn_wmma_f32_16x16x4_f32(
            false, a0, false, b0, (short)0, acc, false, false);
        if (KK > 4) {
            v2f b1;
            b1.x = ldmask<KK>(c, k0 + 4, J + lc);
            b1.y = ldmask<KK>(c, k1 + 4, J + lc);
            acc = __builtin_amdgcn_wmma_f32_16x16x4_f32(
                false, a1, false, b1, (short)0, acc, false, false);
        }

        // D layout: VGPR v -> row I + v + 8*half, col J + lc.
        // Stream non-temporally: 524 MB output > 192 MB L2; keep L2 for inputs.
        #pragma unroll
        for (int v = 0; v < 8; ++v) {
            const int row = I + v + (half << 3);
            __builtin_nontemporal_store(acc[v] * scale,
                                        &ob[(size_t)row * 128 + J + lc]);
        }
    }
}

__global__ void __launch_bounds__(256)
power_spectrum_wmma_kernel(const float* __restrict__ c0,
                           const float* __restrict__ c1,
                           const float* __restrict__ c2,
                           const float* __restrict__ c3,
                           float* __restrict__ out)
{
    const int sample = blockIdx.x;          // s*N + n
    const int wave   = threadIdx.x >> 5;    // 0..7 -> tile row tI = wave
    const int lane   = threadIdx.x & 31;
    const int half   = lane >> 4;
    const int lc     = lane & 15;
    const int I      = wave << 4;

    float* outs = out + (size_t)sample * 4 * 128 * 128;

    // Fully static per-l dispatch: static base pointers (global_load + saddr),
    // compile-time K masks, immediate scale factors.
    gram_tile_row<1>(c0 + (size_t)sample * 1 * 128, outs,             1.0f,
                     I, lc, half);
    gram_tile_row<3>(c1 + (size_t)sample * 3 * 128, outs + 16384,     0.57735026918962576f,
                     I, lc, half);
    gram_tile_row<5>(c2 + (size_t)sample * 5 * 128, outs + 2 * 16384, 0.44721359549995794f,
                     I, lc, half);
    gram_tile_row<7>(c3 + (size_t)sample * 7 * 128, outs + 3 * 16384, 0.37796447300922722f,
                     I, lc, half);
}

extern "C" void kernel_launch(void* const* d_in, const int* in_sizes, int n_in,
                              void* d_out, int out_size, void* d_ws, size_t ws_size,
                              hipStream_t stream) {
    const float* c0 = (const float*)d_in[0];
    const float* c1 = (const float*)d_in[1];
    const float* c2 = (const float*)d_in[2];
    const float* c3 = (const float*)d_in[3];
    float* out = (float*)d_out;

    // in_sizes[0] = S * N * 1 * 128 -> total (s,n) pairs
    const int total_sn = in_sizes[0] / 128;

    power_spectrum_wmma_kernel<<<dim3(total_sn), dim3(256), 0, stream>>>(
        c0, c1, c2, c3, out);
}